// Top1Gate_48369921688083
// MI455X (gfx1250) — compile-verified
//
#include <hip/hip_runtime.h>

// Problem constants (from reference)
#define SS 8192     // tokens
#define DD 2048     // model dim
#define EE 64       // experts
#define CC 128      // capacity
#define TILE_M 32   // tokens per workgroup in gate kernel
#define NTILES (SS / TILE_M)   // 256

typedef __attribute__((ext_vector_type(2))) float v2f;
typedef __attribute__((ext_vector_type(4))) float v4f;
typedef __attribute__((ext_vector_type(8))) float v8f;

// ---------------------------------------------------------------------------
// Kernel 1: zero the 536MB output (bandwidth floor of this problem).
// Non-temporal stores: output is write-once streaming data 2.8x the L2 size,
// so keep it out of the cache hierarchy (TH=NT) instead of rinsing L2.
// ---------------------------------------------------------------------------
__global__ void zero_kernel(float* __restrict__ out, long long n) {
    long long i  = (long long)blockIdx.x * blockDim.x + threadIdx.x;
    long long n4 = n >> 2;
    v4f* p = (v4f*)out;
    v4f z = {0.f, 0.f, 0.f, 0.f};
    long long stride = (long long)gridDim.x * blockDim.x;
    for (long long j = i; j < n4; j += stride) {
        __builtin_nontemporal_store(z, p + j);
    }
    if (i == 0) {
        for (long long j = n4 << 2; j < n; ++j) out[j] = 0.f;
    }
}

// ---------------------------------------------------------------------------
// Kernel 2: fp32 WMMA logits GEMM + softmax + argmax + per-tile stats
//   grid = 256 blocks of 64 threads (2 waves); each wave: 16 tokens x 64 experts
// ---------------------------------------------------------------------------
__global__ __launch_bounds__(64)
void gate_kernel(const float* __restrict__ x, const float* __restrict__ wg,
                 int* __restrict__ eid_g, int* __restrict__ rank_g,
                 float* __restrict__ gate_g,
                 int* __restrict__ tile_count, float* __restrict__ tile_gsum) {
    __shared__ float lg[TILE_M][EE];  // logits, then unnormalized exp()
    __shared__ float inv[TILE_M];     // 1/sum(exp)
    __shared__ int   eid[TILE_M];     // argmax expert

    const int tid   = threadIdx.x;
    const int wave  = tid >> 5;
    const int lane  = tid & 31;
    const int mrow  = lane & 15;           // M (or N) index within 16
    const int khalf = (lane >> 4) << 1;    // 0 for lanes 0-15, 2 for lanes 16-31
    const int tokBase = blockIdx.x * TILE_M + wave * 16;

    // A fragment source: x rows (tokens); B fragments: wg rows (experts)
    const float* xa = x  + (size_t)(tokBase + mrow) * DD + khalf;
    const float* w0 = wg + (size_t)( 0 + mrow) * DD + khalf;
    const float* w1 = wg + (size_t)(16 + mrow) * DD + khalf;
    const float* w2 = wg + (size_t)(32 + mrow) * DD + khalf;
    const float* w3 = wg + (size_t)(48 + mrow) * DD + khalf;

    v8f acc0 = {}, acc1 = {}, acc2 = {}, acc3 = {};

    for (int k0 = 0; k0 < DD; k0 += 4) {
        v2f a  = *(const v2f*)(xa + k0);
        v2f b0 = *(const v2f*)(w0 + k0);
        v2f b1 = *(const v2f*)(w1 + k0);
        v2f b2 = *(const v2f*)(w2 + k0);
        v2f b3 = *(const v2f*)(w3 + k0);
        acc0 = __builtin_amdgcn_wmma_f32_16x16x4_f32(false, a, false, b0, (short)0, acc0, false, false);
        acc1 = __builtin_amdgcn_wmma_f32_16x16x4_f32(false, a, false, b1, (short)0, acc1, false, false);
        acc2 = __builtin_amdgcn_wmma_f32_16x16x4_f32(false, a, false, b2, (short)0, acc2, false, false);
        acc3 = __builtin_amdgcn_wmma_f32_16x16x4_f32(false, a, false, b3, (short)0, acc3, false, false);
    }

    // C/D layout: VGPR j -> M=j (lanes 0-15) or M=j+8 (lanes 16-31), N=lane&15
    {
        const int rbase = wave * 16 + ((lane >> 4) << 3);
        const int c     = lane & 15;
        #pragma unroll
        for (int j = 0; j < 8; ++j) {
            lg[rbase + j][ 0 + c] = acc0[j];
            lg[rbase + j][16 + c] = acc1[j];
            lg[rbase + j][32 + c] = acc2[j];
            lg[rbase + j][48 + c] = acc3[j];
        }
    }
    __syncthreads();

    // Per-token softmax + argmax (threads 0..31 each own a token row)
    if (tid < TILE_M) {
        const int t = tid;
        float mx = lg[t][0]; int am = 0;
        #pragma unroll 8
        for (int e = 1; e < EE; ++e) {
            float v = lg[t][e];
            if (v > mx) { mx = v; am = e; }  // first max wins (matches argmax)
        }
        float sum = 0.f;
        #pragma unroll 8
        for (int e = 0; e < EE; ++e) {
            float g = __expf(lg[t][e] - mx);
            lg[t][e] = g;            // keep unnormalized exp in LDS
            sum += g;
        }
        float is = 1.0f / sum;       // softmax value at argmax = gates1_s
        inv[t] = is;
        eid[t] = am;
        const int s = blockIdx.x * TILE_M + t;
        eid_g[s]  = am;
        gate_g[s] = is;
    }
    __syncthreads();

    // Rank within tile (deterministic, token order)
    if (tid < TILE_M) {
        const int t = tid;
        const int my = eid[t];
        int r = 0;
        for (int t2 = 0; t2 < t; ++t2) r += (eid[t2] == my) ? 1 : 0;
        rank_g[blockIdx.x * TILE_M + t] = r;
    }

    // Per-expert tile stats (all 64 threads, fixed order -> deterministic)
    {
        const int e = tid;
        int cnt = 0;
        float gs = 0.f;
        for (int t = 0; t < TILE_M; ++t) {
            gs  += lg[t][e] * inv[t];          // normalized gate for (t,e)
            cnt += (eid[t] == e) ? 1 : 0;
        }
        tile_count[blockIdx.x * EE + e] = cnt;
        tile_gsum [blockIdx.x * EE + e] = gs;
    }
}

// ---------------------------------------------------------------------------
// Kernel 3: exclusive scan of tile histograms per expert + l_aux
// ---------------------------------------------------------------------------
__global__ __launch_bounds__(64)
void scan_kernel(const int* __restrict__ tile_count,
                 const float* __restrict__ tile_gsum,
                 int* __restrict__ tile_offset, float* __restrict__ out0) {
    __shared__ float red[EE];
    const int e = threadIdx.x;
    int run = 0;
    float gs = 0.f;
    for (int t = 0; t < NTILES; ++t) {
        tile_offset[t * EE + e] = run;
        run += tile_count[t * EE + e];
        gs  += tile_gsum [t * EE + e];
    }
    // l_aux = mean_e(me*ce)*E^2 = (E/S^2) * sum_e gsum_e * cnt_e
    red[e] = gs * (float)run;
    __syncthreads();
    for (int s = 32; s > 0; s >>= 1) {
        if (e < s) red[e] += red[e + s];
        __syncthreads();
    }
    if (e == 0) out0[0] = red[0] * ((float)EE / ((float)SS * (float)SS));
}

// ---------------------------------------------------------------------------
// Kernel 4: scatter nonzeros of combine1_sec and dispatch_mask
// ---------------------------------------------------------------------------
__global__ void scatter_kernel(const int* __restrict__ eid_g,
                               const int* __restrict__ rank_g,
                               const float* __restrict__ gate_g,
                               const int* __restrict__ tile_offset,
                               float* __restrict__ out) {
    const int s = blockIdx.x * blockDim.x + threadIdx.x;
    if (s >= SS) return;
    const int e   = eid_g[s];
    const int loc = tile_offset[(s / TILE_M) * EE + e] + rank_g[s];
    if (loc < CC) {
        const size_t base = 1 + (size_t)s * (EE * CC) + (size_t)e * CC + (size_t)loc;
        out[base] = gate_g[s];                          // combine1_sec
        out[base + (size_t)SS * EE * CC] = 1.0f;        // dispatch_mask (as float)
    }
}

// ---------------------------------------------------------------------------
extern "C" void kernel_launch(void* const* d_in, const int* in_sizes, int n_in,
                              void* d_out, int out_size, void* d_ws, size_t ws_size,
                              hipStream_t stream) {
    const float* x  = (const float*)d_in[0];   // [S, D]
    const float* wg = (const float*)d_in[1];   // [E, D]
    float* out = (float*)d_out;                // [1 + S*E*C + S*E*C]

    // Workspace layout
    int*   eid_g       = (int*)d_ws;                       // S
    int*   rank_g      = eid_g + SS;                       // S
    float* gate_g      = (float*)(rank_g + SS);            // S
    int*   tile_count  = (int*)(gate_g + SS);              // NTILES*E
    int*   tile_offset = tile_count + NTILES * EE;         // NTILES*E
    float* tile_gsum   = (float*)(tile_offset + NTILES * EE); // NTILES*E

    zero_kernel<<<4096, 256, 0, stream>>>(out, (long long)out_size);
    gate_kernel<<<NTILES, 64, 0, stream>>>(x, wg, eid_g, rank_g, gate_g,
                                           tile_count, tile_gsum);
    scan_kernel<<<1, 64, 0, stream>>>(tile_count, tile_gsum, tile_offset, out);
    scatter_kernel<<<SS / 256, 256, 0, stream>>>(eid_g, rank_g, gate_g,
                                                 tile_offset, out);
}